// CrossMamba_34522947125651
// MI455X (gfx1250) — compile-verified
//
#include <hip/hip_runtime.h>
#include <hip/hip_bf16.h>
#include <math.h>

// ---------------- problem constants (match reference) ----------------
constexpr int B_SZ    = 4;
constexpr int C_DIM   = 256;          // d_model
constexpr int L_SEQ   = 64 * 64;      // 4096
constexpr int D_INNER = 512;
constexpr int D_STATE = 16;
constexpr int D_CONV  = 4;
constexpr int DT_RANK = 16;
constexpr int XZ_DIM  = 2 * D_INNER;  // 1024
constexpr int XDBL    = DT_RANK + 2 * D_STATE; // 48
constexpr int M_TOT   = B_SZ * L_SEQ; // 16384 rows (b,l) flattened

// ---------------- WMMA fp32 16x16x4 ----------------
typedef __attribute__((ext_vector_type(2))) float v2f;
typedef __attribute__((ext_vector_type(8))) float v8f;

__device__ __forceinline__ v8f wmma_f32(v2f a, v2f b, v8f c) {
  // 8 args: (neg_a, A, neg_b, B, c_mod, C, reuse_a, reuse_b)
  return __builtin_amdgcn_wmma_f32_16x16x4_f32(false, a, false, b, (short)0, c,
                                               false, false);
}

// Fragment index helpers (per CDNA5 ISA §7.12.2):
//  A (16x4, MxK):  m = lane&15,  k = 2*(lane>>4) + vgpr{0,1}
//  B (4x16, KxN):  k = 2*(lane>>4) + vgpr{0,1},  n = lane&15
//  C/D (16x16):    element in vgpr r: m = r + 8*(lane>>4), n = lane&15

__device__ __forceinline__ float silu_f(float v) {
  return v / (1.0f + __expf(-v));
}

// ---------------- Kernel 1: in_proj GEMM ----------------
// xz[b,l,d] = sum_c x[b,c,l] * W[d,c];  M=16384, N=1024, K=256
// one wave -> 16(M) x 64(N) tile (4 accumulators, A reused)
__global__ void k_inproj(const float* __restrict__ x,
                         const float* __restrict__ W,
                         float* __restrict__ xz) {
  const int wid  = (blockIdx.x * blockDim.x + threadIdx.x) >> 5;
  const int lane = threadIdx.x & 31;
  constexpr int NG = XZ_DIM / 64; // 16 n-groups
  const int mtile = wid / NG;
  const int n0    = (wid % NG) * 64;
  const int m0    = mtile * 16;

  const int mrow = m0 + (lane & 15);      // A row for this lane
  const int bb   = mrow >> 12;            // l rows never cross b (4096%16==0)
  const int ll   = mrow & (L_SEQ - 1);
  const int kl   = 2 * (lane >> 4);
  const int nl   = lane & 15;

  v8f acc[4] = {};
  for (int kc = 0; kc < C_DIM; kc += 16) {
    // unconditional speculative prefetch of next k-panel (clamped, branch-free)
    const int kp = (kc + 16 < C_DIM) ? (kc + 16) : kc;
    __builtin_prefetch(&x[(bb * C_DIM + kp + kl) * L_SEQ + ll], 0, 3);
    __builtin_prefetch(&W[(n0 + nl) * C_DIM + kp], 0, 3);
#pragma unroll
    for (int k4 = 0; k4 < 16; k4 += 4) {
      const int k0 = kc + k4;
      v2f a;
      a.x = x[(bb * C_DIM + k0 + kl + 0) * L_SEQ + ll];
      a.y = x[(bb * C_DIM + k0 + kl + 1) * L_SEQ + ll];
#pragma unroll
      for (int t = 0; t < 4; ++t) {
        const int n = n0 + 16 * t + nl;
        v2f bf;
        bf.x = W[n * C_DIM + k0 + kl + 0];
        bf.y = W[n * C_DIM + k0 + kl + 1];
        acc[t] = wmma_f32(a, bf, acc[t]);
      }
    }
  }
#pragma unroll
  for (int t = 0; t < 4; ++t) {
    const int n = n0 + 16 * t + nl;
#pragma unroll
    for (int r = 0; r < 8; ++r) {
      const int mm = m0 + r + 8 * (lane >> 4);
      xz[mm * XZ_DIM + n] = acc[t][r];
    }
  }
}

// ---------------- Kernel 2: causal depthwise conv1d + SiLU ----------------
// u[b,l,d] = silu(cb[d] + sum_j cw[d,j] * xi[b, l-3+j, d]), xi = xz[..., :512]
__global__ void k_conv(const float* __restrict__ xz,
                       const float* __restrict__ cw,
                       const float* __restrict__ cb,
                       float* __restrict__ u) {
  const int idx = blockIdx.x * blockDim.x + threadIdx.x;
  if (idx >= M_TOT * D_INNER) return;
  const int d = idx & (D_INNER - 1);
  const int l = (idx >> 9) & (L_SEQ - 1);
  const int b = idx >> 21;
  float s = cb[d];
#pragma unroll
  for (int j = 0; j < D_CONV; ++j) {
    const int ls = l - (D_CONV - 1) + j;
    if (ls >= 0) s += cw[d * D_CONV + j] * xz[((b << 12) + ls) * XZ_DIM + d];
  }
  u[idx] = silu_f(s);
}

// ---------------- Kernel 3: x_proj GEMM ----------------
// xd[m,e] = sum_d u[m,d] * W[e,d];  M=16384, N=48, K=512; wave -> 16x48
__global__ void k_xproj(const float* __restrict__ u,
                        const float* __restrict__ W,
                        float* __restrict__ xd) {
  const int wid  = (blockIdx.x * blockDim.x + threadIdx.x) >> 5;
  const int lane = threadIdx.x & 31;
  const int m0   = wid * 16;
  const int kl   = 2 * (lane >> 4);
  const int nl   = lane & 15;
  const int mrow = m0 + nl;

  v8f acc[3] = {};
  for (int kc = 0; kc < D_INNER; kc += 16) {
    const int kp = (kc + 16 < D_INNER) ? (kc + 16) : kc;
    __builtin_prefetch(&u[mrow * D_INNER + kp], 0, 3);
    __builtin_prefetch(&W[nl * D_INNER + kp], 0, 3);
#pragma unroll
    for (int k4 = 0; k4 < 16; k4 += 4) {
      const int k0 = kc + k4;
      v2f a;
      a.x = u[mrow * D_INNER + k0 + kl + 0];
      a.y = u[mrow * D_INNER + k0 + kl + 1];
#pragma unroll
      for (int t = 0; t < 3; ++t) {
        const int n = 16 * t + nl;
        v2f bf;
        bf.x = W[n * D_INNER + k0 + kl + 0];
        bf.y = W[n * D_INNER + k0 + kl + 1];
        acc[t] = wmma_f32(a, bf, acc[t]);
      }
    }
  }
#pragma unroll
  for (int t = 0; t < 3; ++t) {
    const int n = 16 * t + nl;
#pragma unroll
    for (int r = 0; r < 8; ++r) {
      const int mm = m0 + r + 8 * (lane >> 4);
      xd[mm * XDBL + n] = acc[t][r];
    }
  }
}

// ---------------- Kernel 4: dt_proj GEMM + bias + softplus ----------------
// dt[m,d] = softplus(sum_r xd[m,r] * W[d,r] + bias[d]); M=16384, N=512, K=16
__global__ void k_dtproj(const float* __restrict__ xd,
                         const float* __restrict__ W,
                         const float* __restrict__ bias,
                         float* __restrict__ dt) {
  const int wid  = (blockIdx.x * blockDim.x + threadIdx.x) >> 5;
  const int lane = threadIdx.x & 31;
  constexpr int NG = D_INNER / 64; // 8
  const int m0 = (wid / NG) * 16;
  const int n0 = (wid % NG) * 64;
  const int kl = 2 * (lane >> 4);
  const int nl = lane & 15;
  const int mrow = m0 + nl;

  v8f acc[4] = {};
#pragma unroll
  for (int k0 = 0; k0 < DT_RANK; k0 += 4) {
    v2f a;
    a.x = xd[mrow * XDBL + k0 + kl + 0];
    a.y = xd[mrow * XDBL + k0 + kl + 1];
#pragma unroll
    for (int t = 0; t < 4; ++t) {
      const int n = n0 + 16 * t + nl;
      v2f bf;
      bf.x = W[n * DT_RANK + k0 + kl + 0];
      bf.y = W[n * DT_RANK + k0 + kl + 1];
      acc[t] = wmma_f32(a, bf, acc[t]);
    }
  }
#pragma unroll
  for (int t = 0; t < 4; ++t) {
    const int n = n0 + 16 * t + nl;
    const float bv = bias[n];
#pragma unroll
    for (int r = 0; r < 8; ++r) {
      const int mm = m0 + r + 8 * (lane >> 4);
      float v = acc[t][r] + bv;
      // numerically stable softplus
      v = (v > 20.0f) ? v : log1pf(__expf(v));
      dt[mm * D_INNER + n] = v;
    }
  }
}

// ---------------- Kernel 5: selective scan (sequential in L) ----------------
// One wave handles 2 sequences (b,d); lane = (seq_half, n=state).
// Software-pipelined: step l+1's operands are in flight while step l's
// exp+fma+shuffle chain executes, so only the recurrence itself is serial.
// h = exp(dt*A)*h + (dt*u)*B ; y = <h,C> ; y = (y + u*D)*silu(z) over u.
__global__ void k_scan(const float* __restrict__ xz,
                       const float* __restrict__ dt,
                       const float* __restrict__ xd,
                       const float* __restrict__ A_log,
                       const float* __restrict__ Dp,
                       float* __restrict__ u /* in: u, out: gated y */) {
  const int wid  = (blockIdx.x * blockDim.x + threadIdx.x) >> 5;
  const int lane = threadIdx.x & 31;
  const int seq  = wid * 2 + (lane >> 4);   // seq = b*512 + d, 2048 total
  const int n    = lane & 15;
  const int b    = seq >> 9;                // both half-waves share b
  const int d    = seq & (D_INNER - 1);

  const float A  = -__expf(A_log[d * D_STATE + n]);
  const float Dv = Dp[d];
  const int base = (b << 12);

  // preload step 0
  float uv  = u[base * D_INNER + d];
  float dtv = dt[base * D_INNER + d];
  float Bv  = xd[base * XDBL + DT_RANK + n];
  float Cv  = xd[base * XDBL + DT_RANK + D_STATE + n];

  float h = 0.0f;
  for (int l = 0; l < L_SEQ; ++l) {
    // issue next step's loads first (clamped at the end; value unused there)
    const int ln   = (l + 1 < L_SEQ) ? (l + 1) : l;
    const int rown = base + ln;
    const float uv_n  = u[rown * D_INNER + d];
    const float dtv_n = dt[rown * D_INNER + d];
    const float Bv_n  = xd[rown * XDBL + DT_RANK + n];
    const float Cv_n  = xd[rown * XDBL + DT_RANK + D_STATE + n];

    // recurrence (critical path: exp + 2 fma + reduction)
    h = __expf(dtv * A) * h + (dtv * uv) * Bv;

    float p = h * Cv;                 // reduce over the 16 state lanes
    p += __shfl_xor(p, 1, 32);
    p += __shfl_xor(p, 2, 32);
    p += __shfl_xor(p, 4, 32);
    p += __shfl_xor(p, 8, 32);

    if (n == 0) {
      const int row = base + l;
      const float zv = xz[row * XZ_DIM + D_INNER + d];
      u[row * D_INNER + d] = (p + uv * Dv) * silu_f(zv);
    }

    uv = uv_n; dtv = dtv_n; Bv = Bv_n; Cv = Cv_n;
  }
}

// ---------------- Kernel 6: out_proj GEMM + transpose store ----------------
// out[b,c,l] = sum_d y[b,l,d] * W[c,d]; M=16384, N=256, K=512; wave -> 16x64
__global__ void k_outproj(const float* __restrict__ y,
                          const float* __restrict__ W,
                          float* __restrict__ out) {
  const int wid  = (blockIdx.x * blockDim.x + threadIdx.x) >> 5;
  const int lane = threadIdx.x & 31;
  constexpr int NG = C_DIM / 64; // 4
  const int m0 = (wid / NG) * 16;
  const int n0 = (wid % NG) * 64;
  const int kl = 2 * (lane >> 4);
  const int nl = lane & 15;
  const int mrow = m0 + nl;

  v8f acc[4] = {};
  for (int kc = 0; kc < D_INNER; kc += 16) {
    const int kp = (kc + 16 < D_INNER) ? (kc + 16) : kc;
    __builtin_prefetch(&y[mrow * D_INNER + kp], 0, 3);
    __builtin_prefetch(&W[(n0 + nl) * D_INNER + kp], 0, 3);
#pragma unroll
    for (int k4 = 0; k4 < 16; k4 += 4) {
      const int k0 = kc + k4;
      v2f a;
      a.x = y[mrow * D_INNER + k0 + kl + 0];
      a.y = y[mrow * D_INNER + k0 + kl + 1];
#pragma unroll
      for (int t = 0; t < 4; ++t) {
        const int n = n0 + 16 * t + nl;
        v2f bf;
        bf.x = W[n * D_INNER + k0 + kl + 0];
        bf.y = W[n * D_INNER + k0 + kl + 1];
        acc[t] = wmma_f32(a, bf, acc[t]);
      }
    }
  }
#pragma unroll
  for (int t = 0; t < 4; ++t) {
    const int c = n0 + 16 * t + nl;
#pragma unroll
    for (int r = 0; r < 8; ++r) {
      const int mm = m0 + r + 8 * (lane >> 4);
      const int bb = mm >> 12;
      const int ll = mm & (L_SEQ - 1);
      out[(bb * C_DIM + c) * L_SEQ + ll] = acc[t][r];
    }
  }
}

// ---------------- host-side launch ----------------
extern "C" void kernel_launch(void* const* d_in, const int* in_sizes, int n_in,
                              void* d_out, int out_size, void* d_ws,
                              size_t ws_size, hipStream_t stream) {
  (void)in_sizes; (void)n_in; (void)out_size; (void)ws_size;

  const float* x         = (const float*)d_in[0];
  const float* in_proj_w = (const float*)d_in[1];
  const float* conv_w    = (const float*)d_in[2];
  const float* conv_b    = (const float*)d_in[3];
  const float* x_proj_w  = (const float*)d_in[4];
  const float* dt_proj_w = (const float*)d_in[5];
  const float* dt_proj_b = (const float*)d_in[6];
  const float* A_log     = (const float*)d_in[7];
  const float* Dp        = (const float*)d_in[8];
  const float* out_proj_w= (const float*)d_in[9];
  float* out = (float*)d_out;

  // workspace layout (floats): xz | u(->y) | x_dbl | dt  (~137 MB)
  float* ws = (float*)d_ws;
  float* xz = ws;
  float* u  = xz + (size_t)M_TOT * XZ_DIM;
  float* xd = u  + (size_t)M_TOT * D_INNER;
  float* dt = xd + (size_t)M_TOT * XDBL;

  // 1) in_proj: 1024 m-tiles * 16 n-groups = 16384 waves
  k_inproj<<<(16384 * 32) / 256, 256, 0, stream>>>(x, in_proj_w, xz);
  // 2) conv + silu: 8.4M threads
  k_conv<<<(M_TOT * D_INNER + 255) / 256, 256, 0, stream>>>(xz, conv_w, conv_b, u);
  // 3) x_proj: 1024 waves
  k_xproj<<<(1024 * 32) / 256, 256, 0, stream>>>(u, x_proj_w, xd);
  // 4) dt_proj + softplus: 1024 * 8 = 8192 waves
  k_dtproj<<<(8192 * 32) / 256, 256, 0, stream>>>(xd, dt_proj_w, dt_proj_b, dt);
  // 5) selective scan: 2048 sequences / 2 per wave = 1024 waves
  k_scan<<<(1024 * 32) / 256, 256, 0, stream>>>(xz, dt, xd, A_log, Dp, u);
  // 6) out_proj: 1024 * 4 = 4096 waves
  k_outproj<<<(4096 * 32) / 256, 256, 0, stream>>>(u, out_proj_w, out);
}